// MPNNBlock_14121852469811
// MI455X (gfx1250) — compile-verified
//
#include <hip/hip_runtime.h>
#include <hip/hip_bf16.h>

typedef float v2f __attribute__((ext_vector_type(2)));
typedef float v8f __attribute__((ext_vector_type(8)));

#define H   128
#define ED  32
#define K1E (2 * H + ED)   // 288 edge-MLP layer1 K
#define K1N (2 * H)        // 256 node-MLP layer1 K
#define LN_EPS 1e-5f

__device__ __forceinline__ v8f v8zero() {
    v8f z = {0.f, 0.f, 0.f, 0.f, 0.f, 0.f, 0.f, 0.f};
    return z;
}

__device__ __forceinline__ v8f wmma_f32(v2f a, v2f b, v8f c) {
    // D = A(16x4) * B(4x16) + C(16x16), full fp32
    return __builtin_amdgcn_wmma_f32_16x16x4_f32(
        /*neg_a=*/false, a, /*neg_b=*/false, b,
        /*c_mod=*/(short)0, c, /*reuse_a=*/false, /*reuse_b=*/false);
}

// One 4-wide K step: A fragment from *aptr, 8 B fragments from bptr (paired-K LDS),
// accumulate into 8 16x16 tiles. Loads all B frags first -> single dscnt wait per step.
__device__ __forceinline__ void kstep(const float* aptr, const float* bptr, v8f acc[8]) {
    v2f a = *(const v2f*)aptr;
    v2f b[8];
#pragma unroll
    for (int n = 0; n < 8; ++n) b[n] = *(const v2f*)(bptr + n * 32);
#pragma unroll
    for (int n = 0; n < 8; ++n) acc[n] = wmma_f32(a, b[n], acc[n]);
}

__device__ __forceinline__ float gelu_exact(float v) {
    return 0.5f * v * (1.0f + erff(v * 0.70710678118654752f));
}

// ---------------------------------------------------------------- init
__global__ void mpnn_init_kernel(float* __restrict__ ws, long long total) {
    long long i = (long long)blockIdx.x * blockDim.x + threadIdx.x;
    long long stride = (long long)gridDim.x * blockDim.x;
    for (; i < total; i += stride) ws[i] = 0.0f;
}

// ---------------------------------------------------------------- edge MLP + scatter
// LDS: eW1 paired (288*128) | eW2 paired (128*128) | eb1 (128) | eb2 (128) | h stage (8*16*128)
__global__ void __launch_bounds__(256, 1)
mpnn_edge_kernel(const float* __restrict__ x,
                 const int* __restrict__ eidx,     // [2, E] (src row, dst row)
                 const float* __restrict__ ea,     // [E, 32]
                 const float* __restrict__ eW1,    // [288, 128]
                 const float* __restrict__ eb1,
                 const float* __restrict__ eW2,    // [128, 128]
                 const float* __restrict__ eb2,
                 float* __restrict__ agg,          // [N, 128]
                 float* __restrict__ cnt,          // [N]
                 int E) {
    extern __shared__ float smem[];
    float* sW1 = smem;               // paired-K: pos(k,n) = (k>>1)*256 + n*2 + (k&1)
    float* sW2 = sW1 + K1E * H;
    float* sB1 = sW2 + H * H;
    float* sB2 = sB1 + H;
    float* sH  = sB2 + H;            // 8 waves * 16*128

    for (int i = threadIdx.x; i < K1E * H; i += blockDim.x) {
        int k = i / H, n = i - k * H;
        sW1[(k >> 1) * (2 * H) + n * 2 + (k & 1)] = eW1[i];
    }
    for (int i = threadIdx.x; i < H * H; i += blockDim.x) {
        int k = i / H, n = i - k * H;
        sW2[(k >> 1) * (2 * H) + n * 2 + (k & 1)] = eW2[i];
    }
    for (int i = threadIdx.x; i < H; i += blockDim.x) { sB1[i] = eb1[i]; sB2[i] = eb2[i]; }
    __syncthreads();

    const int lane = threadIdx.x & 31;
    const int wave = threadIdx.x >> 5;
    const int half = lane >> 4;      // selects K pair within a 4-wide K step
    const int mrow = lane & 15;      // A-fragment row / B-fragment col lane id
    float* hbuf = sH + wave * (16 * H);

    const int ntiles = E >> 4;                                   // E multiple of 16
    const int nwaves = (gridDim.x * blockDim.x) >> 5;
    int gwave = ((int)(blockIdx.x * blockDim.x + threadIdx.x)) >> 5;

    const int BSTEP = 2 * (2 * H);   // paired-K B stride per 4-wide K step (floats)

    for (int tile = gwave; tile < ntiles; tile += nwaves) {
        const int e0 = tile << 4;
        const int e  = e0 + mrow;
        const int srcI = eidx[e];
        const int dstI = eidx[E + e];

        // ---- layer 1: 16x288 @ 288x128, three branch-free K regions ----
        v8f acc[8];
#pragma unroll
        for (int n = 0; n < 8; ++n) acc[n] = v8zero();

        const float* bptr = sW1 + half * (2 * H) + mrow * 2;

        const float* aptr = x + (size_t)srcI * H + half * 2;     // K = [0,128): x[src]
#pragma unroll 4
        for (int k0 = 0; k0 < H; k0 += 4) {
            kstep(aptr, bptr, acc);
            aptr += 4; bptr += BSTEP;
        }
        aptr = x + (size_t)dstI * H + half * 2;                  // K = [128,256): x[dst]
#pragma unroll 4
        for (int k0 = 0; k0 < H; k0 += 4) {
            kstep(aptr, bptr, acc);
            aptr += 4; bptr += BSTEP;
        }
        aptr = ea + (size_t)e * ED + half * 2;                   // K = [256,288): edge_attr
#pragma unroll
        for (int k0 = 0; k0 < ED; k0 += 4) {
            kstep(aptr, bptr, acc);
            aptr += 4; bptr += BSTEP;
        }

        // bias + exact GELU, stage h (16x128) in per-wave LDS (C-layout -> row-major)
#pragma unroll
        for (int n = 0; n < 8; ++n) {
#pragma unroll
            for (int v = 0; v < 8; ++v) {
                int row = v + half * 8;
                int col = n * 16 + mrow;
                hbuf[row * H + col] = gelu_exact(acc[n][v] + sB1[col]);
            }
        }

        // ---- layer 2: 16x128 @ 128x128 ----
        v8f msg[8];
#pragma unroll
        for (int n = 0; n < 8; ++n) msg[n] = v8zero();

        bptr = sW2 + half * (2 * H) + mrow * 2;
        aptr = hbuf + mrow * H + half * 2;
#pragma unroll 4
        for (int k0 = 0; k0 < H; k0 += 4) {
            kstep(aptr, bptr, msg);
            aptr += 4; bptr += BSTEP;
        }

        // scatter-add messages into agg[dst]; dst of row fetched cross-lane
#pragma unroll
        for (int v = 0; v < 8; ++v) {
            int row = v + half * 8;
            int d = __shfl(dstI, row, 32);
#pragma unroll
            for (int n = 0; n < 8; ++n) {
                int col = n * 16 + mrow;
                atomicAdd(agg + (size_t)d * H + col, msg[n][v] + sB2[col]);
            }
        }
        if (lane < 16) atomicAdd(cnt + dstI, 1.0f);
    }
}

// ---------------------------------------------------------------- node MLP + residual + LN
// LDS: nW1 paired (256*128) | nW2 paired (128*128) | nb1|nb2|ln_g|ln_b (4*128) | stage (8*16*128)
__global__ void __launch_bounds__(256, 1)
mpnn_node_kernel(const float* __restrict__ x,
                 const float* __restrict__ agg,
                 const float* __restrict__ cnt,
                 const float* __restrict__ nW1,    // [256, 128]
                 const float* __restrict__ nb1,
                 const float* __restrict__ nW2,    // [128, 128]
                 const float* __restrict__ nb2,
                 const float* __restrict__ ln_g,
                 const float* __restrict__ ln_b,
                 float* __restrict__ out,
                 int N) {
    extern __shared__ float smem[];
    float* sW1 = smem;
    float* sW2 = sW1 + K1N * H;
    float* sB1 = sW2 + H * H;
    float* sB2 = sB1 + H;
    float* sG  = sB2 + H;
    float* sBt = sG + H;
    float* sH  = sBt + H;

    for (int i = threadIdx.x; i < K1N * H; i += blockDim.x) {
        int k = i / H, n = i - k * H;
        sW1[(k >> 1) * (2 * H) + n * 2 + (k & 1)] = nW1[i];
    }
    for (int i = threadIdx.x; i < H * H; i += blockDim.x) {
        int k = i / H, n = i - k * H;
        sW2[(k >> 1) * (2 * H) + n * 2 + (k & 1)] = nW2[i];
    }
    for (int i = threadIdx.x; i < H; i += blockDim.x) {
        sB1[i] = nb1[i]; sB2[i] = nb2[i]; sG[i] = ln_g[i]; sBt[i] = ln_b[i];
    }
    __syncthreads();

    const int lane = threadIdx.x & 31;
    const int wave = threadIdx.x >> 5;
    const int half = lane >> 4;
    const int mrow = lane & 15;
    float* hbuf = sH + wave * (16 * H);

    const int ntiles = N >> 4;                                   // N multiple of 16
    const int nwaves = (gridDim.x * blockDim.x) >> 5;
    int gwave = ((int)(blockIdx.x * blockDim.x + threadIdx.x)) >> 5;

    const int BSTEP = 2 * (2 * H);

    for (int tile = gwave; tile < ntiles; tile += nwaves) {
        const int n0 = tile << 4;
        const int node_m = n0 + mrow;                 // row this lane feeds into A fragments
        const float rc = 1.0f / (cnt[node_m] + 1e-8f);

        // ---- layer 1: 16x256 @ 256x128, two branch-free K regions ----
        v8f acc[8];
#pragma unroll
        for (int n = 0; n < 8; ++n) acc[n] = v8zero();

        const float* bptr = sW1 + half * (2 * H) + mrow * 2;

        const float* aptr = x + (size_t)node_m * H + half * 2;   // K = [0,128): x
#pragma unroll 4
        for (int k0 = 0; k0 < H; k0 += 4) {
            kstep(aptr, bptr, acc);
            aptr += 4; bptr += BSTEP;
        }
        aptr = agg + (size_t)node_m * H + half * 2;              // K = [128,256): agg * rc
#pragma unroll 4
        for (int k0 = 0; k0 < H; k0 += 4) {
            v2f a = *(const v2f*)aptr;
            a.x *= rc; a.y *= rc;                                // scatter-mean on the fly
            v2f b[8];
#pragma unroll
            for (int n = 0; n < 8; ++n) b[n] = *(const v2f*)(bptr + n * 32);
#pragma unroll
            for (int n = 0; n < 8; ++n) acc[n] = wmma_f32(a, b[n], acc[n]);
            aptr += 4; bptr += BSTEP;
        }

#pragma unroll
        for (int n = 0; n < 8; ++n) {
#pragma unroll
            for (int v = 0; v < 8; ++v) {
                int row = v + half * 8;
                int col = n * 16 + mrow;
                hbuf[row * H + col] = gelu_exact(acc[n][v] + sB1[col]);
            }
        }

        // ---- layer 2: 16x128 @ 128x128 ----
        v8f o[8];
#pragma unroll
        for (int n = 0; n < 8; ++n) o[n] = v8zero();

        bptr = sW2 + half * (2 * H) + mrow * 2;
        aptr = hbuf + mrow * H + half * 2;
#pragma unroll 4
        for (int k0 = 0; k0 < H; k0 += 4) {
            kstep(aptr, bptr, o);
            aptr += 4; bptr += BSTEP;
        }

        // bias + residual, stage out-tile for layernorm
#pragma unroll
        for (int n = 0; n < 8; ++n) {
#pragma unroll
            for (int v = 0; v < 8; ++v) {
                int row = v + half * 8;
                int col = n * 16 + mrow;
                float ov = o[n][v] + sB2[col] + x[(size_t)(n0 + row) * H + col];
                hbuf[row * H + col] = ov;
            }
        }

        // layernorm: 2 lanes per row, 64 cols each, combine via shfl_xor
        {
            int r  = lane >> 1;
            int c0 = (lane & 1) * 64;
            const float* orow = hbuf + r * H + c0;
            float s1 = 0.f, s2 = 0.f;
#pragma unroll 8
            for (int j = 0; j < 64; ++j) { float v = orow[j]; s1 += v; s2 += v * v; }
            s1 += __shfl_xor(s1, 1, 32);
            s2 += __shfl_xor(s2, 1, 32);
            float mu  = s1 * (1.0f / H);
            float var = s2 * (1.0f / H) - mu * mu;
            float inv = rsqrtf(var + LN_EPS);
            float* op = out + (size_t)(n0 + r) * H + c0;
#pragma unroll 8
            for (int j = 0; j < 64; ++j) {
                int col = c0 + j;
                op[j] = (orow[j] - mu) * inv * sG[col] + sBt[col];
            }
        }
    }
}

// ---------------------------------------------------------------- launch
extern "C" void kernel_launch(void* const* d_in, const int* in_sizes, int n_in,
                              void* d_out, int out_size, void* d_ws, size_t ws_size,
                              hipStream_t stream) {
    const float* x    = (const float*)d_in[0];
    const int*   eidx = (const int*)d_in[1];     // [2, E] int32
    const float* ea   = (const float*)d_in[2];
    const float* eW1  = (const float*)d_in[3];
    const float* eb1  = (const float*)d_in[4];
    const float* eW2  = (const float*)d_in[5];
    const float* eb2  = (const float*)d_in[6];
    const float* nW1  = (const float*)d_in[7];
    const float* nb1  = (const float*)d_in[8];
    const float* nW2  = (const float*)d_in[9];
    const float* nb2  = (const float*)d_in[10];
    const float* lng  = (const float*)d_in[11];
    const float* lnb  = (const float*)d_in[12];

    const int N = in_sizes[0] / H;
    const int E = in_sizes[1] / 2;

    float* agg = (float*)d_ws;               // [N, 128]
    float* cnt = agg + (size_t)N * H;        // [N]
    float* out = (float*)d_out;

    const size_t SMEM_E = (size_t)(K1E * H + H * H + 2 * H + 8 * 16 * H) * sizeof(float); // 279552
    const size_t SMEM_N = (size_t)(K1N * H + H * H + 4 * H + 8 * 16 * H) * sizeof(float); // 264192
    (void)hipFuncSetAttribute((const void*)mpnn_edge_kernel,
                              hipFuncAttributeMaxDynamicSharedMemorySize, (int)SMEM_E);
    (void)hipFuncSetAttribute((const void*)mpnn_node_kernel,
                              hipFuncAttributeMaxDynamicSharedMemorySize, (int)SMEM_N);

    long long total = (long long)N * H + N;
    mpnn_init_kernel<<<1024, 256, 0, stream>>>(agg, total);

    mpnn_edge_kernel<<<512, 256, SMEM_E, stream>>>(x, eidx, ea, eW1, eb1, eW2, eb2,
                                                   agg, cnt, E);

    mpnn_node_kernel<<<256, 256, SMEM_N, stream>>>(x, agg, cnt, nW1, nb1, nW2, nb2,
                                                   lng, lnb, out, N);
}